// STN_61538291417695
// MI455X (gfx1250) — compile-verified
//
#include <hip/hip_runtime.h>

// STN glimpse extraction: separable affine (scale+translate only) bilinear resample.
// Stage 1: row interpolation with fully-coalesced global reads -> LDS (T2).
// Stage 2: column interpolation as dense f32 matmul Out = T2 * Cx^T via
//          V_WMMA_F32_16X16X4_F32 (CDNA5 wave32 WMMA), operands from LDS.

#define H_OBJ   64
#define W_OBJ   64
#define C_IN    3
#define H_IN    256
#define W_IN    256
#define SK      260      // padded K stride (floats): 260 mod 64 == 4 -> conflict-free frag loads
#define THREADS 256      // 8 waves (wave32)

typedef float v2f __attribute__((ext_vector_type(2)));
typedef float v8f __attribute__((ext_vector_type(8)));

__global__ __launch_bounds__(THREADS)
void stn_glimpse_wmma(const float* __restrict__ img,
                      const float* __restrict__ zwhere,
                      const int*   __restrict__ inv_flag,
                      float* __restrict__ out)
{
    extern __shared__ float smem[];
    float* T2 = smem;                        // [64][SK] row-lerped rows (A matrix, M=64,K=256)
    float* Bt = smem + H_OBJ * SK;           // [64][SK] column weights, Bt[col][k] (B^T)

    __shared__ int   s_iy0[H_OBJ], s_iy1[H_OBJ];
    __shared__ float s_wy [H_OBJ];
    __shared__ int   s_ix0[W_OBJ], s_ix1[W_OBJ];
    __shared__ float s_wx [W_OBJ];

    const int n    = blockIdx.x;
    const int t    = threadIdx.x;
    const int lane = t & 31;
    const int wave = t >> 5;
    const int inv  = inv_flag[0];

    const float yc = zwhere[n * 4 + 0] * 2.0f - 1.0f;
    const float xc = zwhere[n * 4 + 1] * 2.0f - 1.0f;
    const float hh = zwhere[n * 4 + 2];
    const float ww = zwhere[n * 4 + 3];

    // ---- per-row / per-column interpolation tables (threads 0..63) ----
    if (t < H_OBJ) {
        const float g = -1.0f + (2.0f / 63.0f) * (float)t;
        // rows (y)
        float sy  = inv ? (g - yc) / hh : g * hh + yc;
        float iy  = fminf(fmaxf((sy + 1.0f) * 0.5f * (float)(H_IN - 1), 0.0f), (float)(H_IN - 1));
        float iyf = floorf(iy);
        s_wy[t]   = iy - iyf;
        int iy0   = (int)iyf;
        iy0       = iy0 < 0 ? 0 : (iy0 > H_IN - 1 ? H_IN - 1 : iy0);
        s_iy0[t]  = iy0;
        s_iy1[t]  = (iy0 + 1 > H_IN - 1) ? H_IN - 1 : iy0 + 1;
        // cols (x)
        float sx  = inv ? (g - xc) / ww : g * ww + xc;
        float ix  = fminf(fmaxf((sx + 1.0f) * 0.5f * (float)(W_IN - 1), 0.0f), (float)(W_IN - 1));
        float ixf = floorf(ix);
        s_wx[t]   = ix - ixf;
        int ix0   = (int)ixf;
        ix0       = ix0 < 0 ? 0 : (ix0 > W_IN - 1 ? W_IN - 1 : ix0);
        s_ix0[t]  = ix0;
        s_ix1[t]  = (ix0 + 1 > W_IN - 1) ? W_IN - 1 : ix0 + 1;
    }

    // ---- build dense column-interp matrix Bt[col][k] (shared by all 3 channels) ----
    for (int i = t; i < W_OBJ * SK; i += THREADS) Bt[i] = 0.0f;
    __syncthreads();
    if (t < W_OBJ) {
        const float wx = s_wx[t];
        Bt[t * SK + s_ix0[t]] += 1.0f - wx;   // ix0==ix1 at the border: weights sum to 1
        Bt[t * SK + s_ix1[t]] += wx;
    }
    __syncthreads();

    // ---- WMMA tile assignment: 16 C-tiles (4x4 of 16x16) over 8 waves, 2 tiles/wave ----
    const int tm  = wave >> 1;                 // tile row (shared by both tiles of this wave)
    const int tn0 = (wave & 1) ? 2 : 0;        // first tile col; second is tn0+1
    const int fm  = lane & 15;                 // row-within-tile for A/B frags, N for D
    const int fk  = (lane >> 4) * 2;           // K sub-offset for A/B frags (ISA 16x4 layout)

    const float* Arow  = T2 + (tm * 16 + fm) * SK + fk;
    const float* Brow0 = Bt + (tn0 * 16 + fm) * SK + fk;
    const float* Brow1 = Brow0 + 16 * SK;

    for (int c = 0; c < C_IN; ++c) {
        const float* imc = img + ((size_t)(n * C_IN + c)) * (H_IN * W_IN);

        // ---- stage 1: row interpolation, fully-coalesced 1024B row reads ----
        for (int r = 0; r < H_OBJ; ++r) {
            const float v0 = imc[(size_t)s_iy0[r] * W_IN + t];
            const float v1 = imc[(size_t)s_iy1[r] * W_IN + t];
            const float wy = s_wy[r];
            T2[r * SK + t] = v0 + wy * (v1 - v0);
        }
        __syncthreads();

        // ---- stage 2: Out(64x64) = T2(64x256) x Bt^T via v_wmma_f32_16x16x4_f32 ----
        v8f acc0 = {0.f, 0.f, 0.f, 0.f, 0.f, 0.f, 0.f, 0.f};
        v8f acc1 = {0.f, 0.f, 0.f, 0.f, 0.f, 0.f, 0.f, 0.f};
#pragma unroll 4
        for (int k0 = 0; k0 < W_IN; k0 += 4) {
            v2f a, b0, b1;
            a.x  = Arow[k0];   a.y  = Arow[k0 + 1];
            b0.x = Brow0[k0];  b0.y = Brow0[k0 + 1];
            b1.x = Brow1[k0];  b1.y = Brow1[k0 + 1];
            acc0 = __builtin_amdgcn_wmma_f32_16x16x4_f32(
                /*neg_a=*/false, a, /*neg_b=*/false, b0,
                /*c_mod=*/(short)0, acc0, /*reuse_a=*/false, /*reuse_b=*/false);
            acc1 = __builtin_amdgcn_wmma_f32_16x16x4_f32(
                false, a, false, b1, (short)0, acc1, false, false);
        }

        // ---- store D tiles: n = lane&15, m = vgpr + 8*(lane>>4) (ISA C/D layout) ----
        float* oc = out + ((size_t)(n * C_IN + c)) * (H_OBJ * W_OBJ);
        const int rbase = tm * 16 + (lane >> 4) * 8;
        const int col0  = tn0 * 16 + fm;
#pragma unroll
        for (int v = 0; v < 8; ++v) {
            oc[(rbase + v) * W_OBJ + col0]      = acc0[v];
            oc[(rbase + v) * W_OBJ + col0 + 16] = acc1[v];
        }
        __syncthreads();   // T2 reused next channel
    }
}

extern "C" void kernel_launch(void* const* d_in, const int* in_sizes, int n_in,
                              void* d_out, int out_size, void* d_ws, size_t ws_size,
                              hipStream_t stream) {
    const float* img = (const float*)d_in[0];   // (512, 3, 256, 256) f32
    const float* zw  = (const float*)d_in[1];   // (512, 4) f32
    const int*   inv = (const int*)d_in[2];     // scalar int
    float*       out = (float*)d_out;           // (512, 12288) f32

    const int N = in_sizes[1] / 4;
    const size_t shmem = (size_t)2 * H_OBJ * SK * sizeof(float);  // ~130 KB dynamic LDS

    (void)hipFuncSetAttribute((const void*)stn_glimpse_wmma,
                              hipFuncAttributeMaxDynamicSharedMemorySize, (int)shmem);

    stn_glimpse_wmma<<<N, THREADS, shmem, stream>>>(img, zw, inv, out);
}